// FraudDetectionGNN_17394617548971
// MI455X (gfx1250) — compile-verified
//
#include <hip/hip_runtime.h>
#include <hip/hip_bf16.h>
#include <math.h>

typedef __attribute__((ext_vector_type(2))) float v2f;
typedef __attribute__((ext_vector_type(8))) float v8f;

#define FIN 165
#define HDIM 128

// ---------------------------------------------------------------------------
// Degree / normalization kernels
// ---------------------------------------------------------------------------
__global__ void zero_i32(int* p, int n) {
    int i = blockIdx.x * blockDim.x + threadIdx.x;
    if (i < n) p[i] = 0;
}

__global__ void count_deg(const int* __restrict__ dst, int* __restrict__ deg, int e) {
    int i = blockIdx.x * blockDim.x + threadIdx.x;
    if (i < e) atomicAdd(&deg[dst[i]], 1);
}

__global__ void calc_dinv(const int* __restrict__ deg, float* __restrict__ dinv, int n) {
    int i = blockIdx.x * blockDim.x + threadIdx.x;
    if (i < n) dinv[i] = rsqrtf((float)deg[i] + 1.0f);  // +1 = self loop
}

// ---------------------------------------------------------------------------
// WMMA f32 GEMM: C[M x 128] = op(A[M x K]) @ B[K x 128], op = optional ReLU.
// One workgroup: 16 rows x 128 cols, 8 waves, each wave one 16x16 tile.
// K staged through LDS in chunks of 32, zero-padded (K=165 case).
// ---------------------------------------------------------------------------
#define BK 32
__global__ __launch_bounds__(256) void gemm_wmma_f32(
    const float* __restrict__ A, const float* __restrict__ B,
    float* __restrict__ C, int M, int K, int lda, int reluA)
{
    __shared__ float As[16][BK + 4];      // +4 pad: conflict-free column reads
    __shared__ float Bs[BK][HDIM + 4];

    const int tid  = threadIdx.x;
    const int wave = tid >> 5;            // 0..7 -> column tile
    const int lane = tid & 31;
    const int l16  = lane & 15;
    const int hi   = lane >> 4;           // K-pair selector (A/B frag layout)
    const int rowBase = blockIdx.x * 16;
    const int colBase = wave * 16;

    v8f acc = {};

    for (int k0 = 0; k0 < K; k0 += BK) {
        // Stage A tile (16 x 32), coalesced along K, zero-fill padding.
        for (int i = tid; i < 16 * BK; i += 256) {
            int r = i / BK, kk = i % BK;
            int gr = rowBase + r, gk = k0 + kk;
            float v = 0.0f;
            if (gr < M && gk < K) {
                v = A[(size_t)gr * lda + gk];
                if (reluA) v = fmaxf(v, 0.0f);
            }
            As[r][kk] = v;
        }
        // Stage B tile (32 x 128), coalesced along columns.
        for (int i = tid; i < BK * HDIM; i += 256) {
            int r = i / HDIM, c = i % HDIM;
            int gk = k0 + r;
            Bs[r][c] = (gk < K) ? B[(size_t)gk * HDIM + c] : 0.0f;
        }
        __syncthreads();

#pragma unroll
        for (int kk = 0; kk < BK; kk += 4) {
            // A frag: lanes 0-15 hold M=0..15 {K=kk,kk+1}; lanes 16-31 {K=kk+2,kk+3}
            v2f a, b;
            a.x = As[l16][kk + hi * 2 + 0];
            a.y = As[l16][kk + hi * 2 + 1];
            // B frag mirrors A: lane%16 = N, lane/16 selects K pair
            b.x = Bs[kk + hi * 2 + 0][colBase + l16];
            b.y = Bs[kk + hi * 2 + 1][colBase + l16];
            acc = __builtin_amdgcn_wmma_f32_16x16x4_f32(
                false, a, false, b, (short)0, acc, false, false);
        }
        __syncthreads();
    }

    // C/D layout: VGPR r, lanes 0-15 -> M=r; lanes 16-31 -> M=8+r; N=lane%16
#pragma unroll
    for (int r = 0; r < 8; r++) {
        int row = rowBase + hi * 8 + r;
        if (row < M) C[(size_t)row * HDIM + colBase + l16] = acc[r];
    }
}

// ---------------------------------------------------------------------------
// agg[i,f] = h[i,f] * dinv[i]^2 + bias[f]   (self-loop term + bias init)
// ---------------------------------------------------------------------------
__global__ __launch_bounds__(256) void agg_init(
    const float* __restrict__ h, const float* __restrict__ dinv,
    const float* __restrict__ bias, float* __restrict__ agg, int n)
{
    long long idx = (long long)blockIdx.x * 256 + threadIdx.x;
    if (idx >= (long long)n * HDIM) return;
    int i = (int)(idx >> 7), f = (int)(idx & 127);
    float di = dinv[i];
    agg[idx] = h[idx] * di * di + bias[f];
}

// ---------------------------------------------------------------------------
// One wave per edge: agg[dst] += h[src] * dinv[src]*dinv[dst]
// Each lane owns a float4 slice of the 128-feature row -> 4 f32 atomics.
// ---------------------------------------------------------------------------
__global__ __launch_bounds__(256) void edge_scatter(
    const int* __restrict__ src, const int* __restrict__ dst,
    const float* __restrict__ dinv, const float* __restrict__ h,
    float* __restrict__ agg, int e)
{
    int wid  = (int)(((long long)blockIdx.x * blockDim.x + threadIdx.x) >> 5);
    int lane = threadIdx.x & 31;
    if (wid >= e) return;
    int s = src[wid], d = dst[wid];
    float norm = dinv[s] * dinv[d];
    const float4* hp = (const float4*)(h + (size_t)s * HDIM);
    float4 v = hp[lane];
    float* ap = agg + (size_t)d * HDIM + lane * 4;
    atomicAdd(ap + 0, v.x * norm);
    atomicAdd(ap + 1, v.y * norm);
    atomicAdd(ap + 2, v.z * norm);
    atomicAdd(ap + 3, v.w * norm);
}

// ---------------------------------------------------------------------------
// out[i, 0:2] = relu(agg[i, :]) @ Wl + bl
// ---------------------------------------------------------------------------
__global__ __launch_bounds__(256) void head_linear(
    const float* __restrict__ agg, const float* __restrict__ Wl,
    const float* __restrict__ bl, float* __restrict__ out, int n)
{
    __shared__ float w[HDIM * 2];
    w[threadIdx.x] = Wl[threadIdx.x];          // 256 == 128*2
    __syncthreads();
    int i = blockIdx.x * 256 + threadIdx.x;
    if (i >= n) return;
    const float4* row = (const float4*)(agg + (size_t)i * HDIM);
    float s0 = 0.0f, s1 = 0.0f;
#pragma unroll
    for (int f4 = 0; f4 < HDIM / 4; f4++) {
        float4 v = row[f4];
        float a0 = fmaxf(v.x, 0.0f), a1 = fmaxf(v.y, 0.0f);
        float a2 = fmaxf(v.z, 0.0f), a3 = fmaxf(v.w, 0.0f);
        int f = f4 * 4;
        s0 += a0 * w[(f + 0) * 2] + a1 * w[(f + 1) * 2] + a2 * w[(f + 2) * 2] + a3 * w[(f + 3) * 2];
        s1 += a0 * w[(f + 0) * 2 + 1] + a1 * w[(f + 1) * 2 + 1] + a2 * w[(f + 2) * 2 + 1] + a3 * w[(f + 3) * 2 + 1];
    }
    out[(size_t)i * 2 + 0] = s0 + bl[0];
    out[(size_t)i * 2 + 1] = s1 + bl[1];
}

// ---------------------------------------------------------------------------
// Launch
// ---------------------------------------------------------------------------
extern "C" void kernel_launch(void* const* d_in, const int* in_sizes, int n_in,
                              void* d_out, int out_size, void* d_ws, size_t ws_size,
                              hipStream_t stream)
{
    const float* x   = (const float*)d_in[0];
    const int*   ei  = (const int*)d_in[1];
    const float* W1  = (const float*)d_in[2];
    const float* b1  = (const float*)d_in[3];
    const float* W2  = (const float*)d_in[4];
    const float* b2  = (const float*)d_in[5];
    const float* Wl  = (const float*)d_in[6];
    const float* bl  = (const float*)d_in[7];
    float* out = (float*)d_out;

    const int N = in_sizes[0] / FIN;
    const int E = in_sizes[1] / 2;
    const int* e_src = ei;
    const int* e_dst = ei + E;

    // Workspace layout
    char* ws = (char*)d_ws;
    int*   deg  = (int*)ws;                                   // N int32
    float* dinv = (float*)(ws + (size_t)N * 4);               // N f32
    size_t off  = ((size_t)N * 8 + 255) & ~(size_t)255;
    float* B0   = (float*)(ws + off);                         // N*128 f32 (h_lin)
    float* B1   = (float*)(ws + off + (size_t)N * HDIM * 4);  // N*128 f32 (agg)

    const int T = 256;
    // 1) degrees -> dinv
    zero_i32<<<(N + T - 1) / T, T, 0, stream>>>(deg, N);
    count_deg<<<(E + T - 1) / T, T, 0, stream>>>(e_dst, deg, E);
    calc_dinv<<<(N + T - 1) / T, T, 0, stream>>>(deg, dinv, N);

    const int gemmBlocks = (N + 15) / 16;
    const long long elems = (long long)N * HDIM;
    const int aggBlocks  = (int)((elems + T - 1) / T);
    const int edgeBlocks = (int)(((long long)E * 32 + T - 1) / T);

    // Layer 1: h = x@W1 ; agg = selfloop+bias ; scatter ; (relu fused next)
    gemm_wmma_f32<<<gemmBlocks, T, 0, stream>>>(x, W1, B0, N, FIN, FIN, 0);
    agg_init<<<aggBlocks, T, 0, stream>>>(B0, dinv, b1, B1, N);
    edge_scatter<<<edgeBlocks, T, 0, stream>>>(e_src, e_dst, dinv, B0, B1, E);

    // Layer 2: h = relu(agg1)@W2 ; agg2 ; scatter
    gemm_wmma_f32<<<gemmBlocks, T, 0, stream>>>(B1, W2, B0, N, HDIM, HDIM, 1);
    agg_init<<<aggBlocks, T, 0, stream>>>(B0, dinv, b2, B1, N);
    edge_scatter<<<edgeBlocks, T, 0, stream>>>(e_src, e_dst, dinv, B0, B1, E);

    // Head: out = relu(agg2) @ Wl + bl
    head_linear<<<(N + T - 1) / T, T, 0, stream>>>(B1, Wl, bl, out, N);
}